// DynamicMemoryInduction_12713103197289
// MI455X (gfx1250) — compile-verified
//
#include <hip/hip_runtime.h>
#include <hip/hip_bf16.h>
#include <math.h>
#include <stdint.h>

// Sizes (fixed by the reference)
#define I_CAPS 8
#define E_DIM  768
#define J_CAPS 8
#define D_DIM  768
#define N_Q    400
#define EPSF   1e-8f

typedef __attribute__((ext_vector_type(2))) float v2f;
typedef __attribute__((ext_vector_type(8))) float v8f;

__device__ __forceinline__ float wsum32(float v) {
#pragma unroll
  for (int off = 16; off > 0; off >>= 1)
    v += __shfl_xor(v, off, 32);
  return v;
}

// ---- CDNA5 async global->LDS copies (ASYNCcnt path) ----------------------
// LDS byte offset: low 32 bits of the flat shared-aperture address
// (ISA 10.2). The instruction's offset immediate
// is added to BOTH the LDS destination and the global address, which matches
// row-linear staging exactly.
__device__ __forceinline__ uint32_t lds_off(const void* p) {
  return (uint32_t)(uintptr_t)p;
}
__device__ __forceinline__ void async_g2l_b128(uint32_t lds, const void* g, int imm) {
  asm volatile("global_load_async_to_lds_b128 %0, %1, off offset:%2"
               :: "v"(lds), "v"(g), "i"(imm) : "memory");
}
__device__ __forceinline__ void async_g2l_b64(uint32_t lds, const void* g) {
  asm volatile("global_load_async_to_lds_b64 %0, %1, off"
               :: "v"(lds), "v"(g) : "memory");
}
__device__ __forceinline__ void wait_async0() {
  asm volatile("s_wait_asynccnt 0x0" ::: "memory");
}

// ---------------------------------------------------------------------------
// Kernel 1: m_proj[i,d] = sum_e m[i,e] * W_n[e,d]          (8x768, tiny)
// ---------------------------------------------------------------------------
__global__ void __launch_bounds__(256)
k_mproj(const float* __restrict__ m, const float* __restrict__ Wn,
        float* __restrict__ mproj) {
  int idx = blockIdx.x * 256 + threadIdx.x;         // 0..6143
  int i = idx / D_DIM, d = idx % D_DIM;
  const float* mr = m + i * E_DIM;
  float acc = 0.f;
#pragma unroll 4
  for (int e = 0; e < E_DIM; ++e)
    acc = fmaf(mr[e], Wn[e * D_DIM + d], acc);
  mproj[idx] = acc;
}

// ---------------------------------------------------------------------------
// Kernel 2: hm[j,i,d] = b[j,i,d] + sum_e W[0,j,0,d,e] * mproj[i,e]
//           (W is broadcast over i in the reference -> use i=0 slice)
// ---------------------------------------------------------------------------
__global__ void __launch_bounds__(256)
k_hatm(const float* __restrict__ W, const float* __restrict__ b,
       const float* __restrict__ mproj, float* __restrict__ hm) {
  int idx = blockIdx.x * 256 + threadIdx.x;         // 0..49151 = (j*8+i)*768+d
  int j = idx / (I_CAPS * D_DIM);
  int rem = idx % (I_CAPS * D_DIM);
  int i = rem / D_DIM, d = rem % D_DIM;
  const float* wr = W + ((size_t)j * I_CAPS * D_DIM + d) * E_DIM;  // W[0,j,0,d,:]
  const float* mp = mproj + i * E_DIM;
  float acc = 0.f;
#pragma unroll 4
  for (int e = 0; e < E_DIM; ++e)
    acc = fmaf(wr[e], mp[e], acc);
  hm[idx] = acc + b[idx];
}

// ---------------------------------------------------------------------------
// Kernel 3: per-(j,i) Pearson stats of hm: mean and sum((x-mean)^2)
// ---------------------------------------------------------------------------
__global__ void __launch_bounds__(256)
k_mstats(const float* __restrict__ hm, float* __restrict__ mstats) {
  int ji = blockIdx.x;                              // 0..63
  const float* x = hm + ji * D_DIM;
  int t = threadIdx.x;
  float s = 0.f, s2 = 0.f;
  for (int d = t; d < D_DIM; d += 256) { float v = x[d]; s += v; s2 = fmaf(v, v, s2); }
  s = wsum32(s); s2 = wsum32(s2);
  __shared__ float r1[8], r2[8];
  if ((t & 31) == 0) { r1[t >> 5] = s; r2[t >> 5] = s2; }
  __syncthreads();
  if (t == 0) {
    float S = 0.f, S2 = 0.f;
#pragma unroll
    for (int k = 0; k < 8; ++k) { S += r1[k]; S2 += r2[k]; }
    float mean = S * (1.f / 768.f);
    mstats[ji]      = mean;                         // mean_m[j,i]
    mstats[64 + ji] = S2 - 768.f * mean * mean;     // ss_m[j,i]
  }
}

// ---------------------------------------------------------------------------
// Kernel 4: the big GEMM, f32 WMMA 16x16x4, double-buffered async LDS fill.
//   hq[n,j,d] = sum_e q[n,e] * W[0,j,0,d,e]
// Block: 256 thr (8 waves). Block tile: M=16 (q rows) x N=256 (d).
// Each wave: two 16x16 accumulators. K staged 32 at a time; the async engine
// fills tile k+1 while WMMA consumes tile k (one wait+barrier per iteration).
// Grid: (d-tile 0..2, m-tile 0..24, j 0..7)
// ---------------------------------------------------------------------------
__global__ void __launch_bounds__(256)
k_gemm_hq(const float* __restrict__ q, const float* __restrict__ W,
          float* __restrict__ hq) {
  const int nt = blockIdx.x;        // 3 tiles of 256 along d
  const int mt = blockIdx.y;        // 25 tiles of 16 along n (25*16 = 400 exact)
  const int j  = blockIdx.z;
  const int t    = threadIdx.x;
  const int wave = t >> 5, lane = t & 31;
  const int r16  = lane & 15, hi = lane >> 4;

  __shared__ float As[2][16][34];   // 16 x 32 (+pad), double buffered
  __shared__ float Bs[2][256][34];  // 256 x 32 (+pad): Bs[buf][d_local][k]

  const float* qbase = q + (size_t)mt * 16 * E_DIM;
  const float* wbase = W + (size_t)j * I_CAPS * D_DIM * E_DIM
                         + (size_t)(nt * 256) * E_DIM;   // W[0,j,0, d0.., :]

  // per-thread staging addresses (fixed across K iterations)
  const int arow = t >> 4;
  const int acol = (t & 15) * 2;
  const uint32_t ldsA[2] = { lds_off(&As[0][arow][acol]), lds_off(&As[1][arow][acol]) };
  const uint32_t ldsB[2] = { lds_off(&Bs[0][t][0]),       lds_off(&Bs[1][t][0]) };
  const float* qrow = qbase + arow * E_DIM + acol;
  const float* wrow = wbase + (size_t)t * E_DIM;

  auto stage = [&](int buf, int k0) {
    async_g2l_b64(ldsA[buf], qrow + k0);
    const float* src = wrow + k0;
#pragma unroll
    for (int u = 0; u < 8; ++u)
      async_g2l_b128(ldsB[buf], src, u * 16);   // LDS delta == global delta
    if (k0 + 32 < E_DIM)                        // L2-warm the tile after next
      __builtin_prefetch(src + 32, 0, 1);
  };

  v8f acc0 = {}; v8f acc1 = {};

  stage(0, 0);
  wait_async0();
  __syncthreads();

  for (int k0 = 0; k0 < E_DIM; k0 += 32) {
    const int cur = (k0 >> 5) & 1;
    if (k0 + 32 < E_DIM)
      stage(cur ^ 1, k0 + 32);                  // overlap fill with WMMA below

#pragma unroll
    for (int kk = 0; kk < 8; ++kk) {
      const int kidx = kk * 4 + hi * 2;  // lanes 0-15: K{0,1}; 16-31: K{2,3}
      v2f a;  a.x  = As[cur][r16][kidx];
              a.y  = As[cur][r16][kidx + 1];
      v2f b0; b0.x = Bs[cur][wave * 32 + r16][kidx];
              b0.y = Bs[cur][wave * 32 + r16][kidx + 1];
      v2f b1; b1.x = Bs[cur][wave * 32 + 16 + r16][kidx];
              b1.y = Bs[cur][wave * 32 + 16 + r16][kidx + 1];
      acc0 = __builtin_amdgcn_wmma_f32_16x16x4_f32(false, a, false, b0,
                                                   (short)0, acc0, false, false);
      acc1 = __builtin_amdgcn_wmma_f32_16x16x4_f32(false, a, false, b1,
                                                   (short)0, acc1, false, false);
    }
    wait_async0();       // next tile's fill complete (this wave)
    __syncthreads();     // all waves: fills done + current tile fully read
  }

  // store: C layout -> VGPR r: M=r (lanes 0-15) / r+8 (lanes 16-31), N = lane%16
  const int dbase = nt * 256 + wave * 32;
#pragma unroll
  for (int r = 0; r < 8; ++r) {
    int mrow = mt * 16 + r + 8 * hi;
    size_t o = ((size_t)mrow * J_CAPS + j) * D_DIM + dbase + r16;
    hq[o]      = acc0[r];
    hq[o + 16] = acc1[r];
  }
}

// ---------------------------------------------------------------------------
// Kernel 5: routing. One block per n. Wave w <-> capsule j (8 waves).
// State exploited: tmp_q independent of i; a[n,j,i,d] = hq*c1 (+ hqt2*c2).
// ---------------------------------------------------------------------------
__global__ void __launch_bounds__(256)
k_routing(const float* __restrict__ hm, const float* __restrict__ mstats,
          const float* __restrict__ hq, float* __restrict__ out) {
  const int n = blockIdx.x;
  const int t = threadIdx.x;
  const int j = t >> 5;             // wave id == capsule j
  const int lane = t & 31;

  __shared__ float sHq[J_CAPS][D_DIM];
  __shared__ float sV1[J_CAPS][D_DIM];
  __shared__ float sV2[J_CAPS][D_DIM];
  __shared__ float sP[64];          // p[j,i]
  __shared__ float sC1[64];         // cos1[j,i]
  __shared__ float sC2[64];         // cos2[j,i]

  // async copy hq[n] (24KB contiguous) into LDS: 6 x b128 per thread,
  // offsets folded into the instruction immediate.
  {
    const char* g = (const char*)(hq + (size_t)n * (J_CAPS * D_DIM)) + t * 16;
    const uint32_t base = lds_off(&sHq[0][0]) + (uint32_t)t * 16u;
#pragma unroll
    for (int c = 0; c < 6; ++c)
      async_g2l_b128(base, g, c * 4096);        // 256 threads * 16B = 4096B
  }
  wait_async0();
  __syncthreads();

  const float* hmj = hm + j * (I_CAPS * D_DIM);

  // ---- p0 = tanh(pearson(hm_ji, hq_j)) ----
  float sum = 0.f, sum2 = 0.f;
  for (int d = lane; d < D_DIM; d += 32) { float v = sHq[j][d]; sum += v; sum2 = fmaf(v, v, sum2); }
  sum = wsum32(sum); sum2 = wsum32(sum2);
  float meanY = sum * (1.f / 768.f);
  float ssY   = sum2 - 768.f * meanY * meanY;
  for (int i = 0; i < I_CAPS; ++i) {
    const float* x = hmj + i * D_DIM;
    float dp = 0.f;
    for (int d = lane; d < D_DIM; d += 32) dp = fmaf(x[d], sHq[j][d], dp);
    dp = wsum32(dp);
    float num = dp - 768.f * mstats[j * 8 + i] * meanY;
    float den = sqrtf(mstats[64 + j * 8 + i] * ssY);
    if (lane == 0) sP[j * 8 + i] = tanhf(num / (den + EPSF));
  }
  __syncthreads();

  // ---- iter 1: hv = sum_i (1/8 + p0) * hm  ->  v1 = squash(hv) ----
  {
    float c[8];
#pragma unroll
    for (int i = 0; i < 8; ++i) c[i] = 0.125f + sP[j * 8 + i];
    float n2 = 0.f;
    for (int d = lane; d < D_DIM; d += 32) {
      float a = 0.f;
#pragma unroll
      for (int i = 0; i < 8; ++i) a = fmaf(c[i], hmj[i * D_DIM + d], a);
      sV1[j][d] = a;
      n2 = fmaf(a, a, n2);
    }
    n2 = wsum32(n2);
    float scale = (n2 / (1.f + n2)) / sqrtf(n2 + EPSF);
    for (int d = lane; d < D_DIM; d += 32) sV1[j][d] *= scale;
  }
  // cos1[j,i] = dot(hm_ji, v1_j)   (wave-local: sV1[j] written by this wave)
  for (int i = 0; i < I_CAPS; ++i) {
    const float* x = hmj + i * D_DIM;
    float dp = 0.f;
    for (int d = lane; d < D_DIM; d += 32) dp = fmaf(x[d], sV1[j][d], dp);
    dp = wsum32(dp);
    if (lane == 0) sC1[j * 8 + i] = dp;
  }
  // p1 with y2 = hqt2 = 0.5*(hq + v1)
  sum = 0.f; sum2 = 0.f;
  for (int d = lane; d < D_DIM; d += 32) {
    float v = 0.5f * (sHq[j][d] + sV1[j][d]); sum += v; sum2 = fmaf(v, v, sum2);
  }
  sum = wsum32(sum); sum2 = wsum32(sum2);
  meanY = sum * (1.f / 768.f); ssY = sum2 - 768.f * meanY * meanY;
  for (int i = 0; i < I_CAPS; ++i) {
    const float* x = hmj + i * D_DIM;
    float dp = 0.f;
    for (int d = lane; d < D_DIM; d += 32)
      dp = fmaf(x[d], 0.5f * (sHq[j][d] + sV1[j][d]), dp);
    dp = wsum32(dp);
    float num = dp - 768.f * mstats[j * 8 + i] * meanY;
    float den = sqrtf(mstats[64 + j * 8 + i] * ssY);
    if (lane == 0) sP[j * 8 + i] = tanhf(num / (den + EPSF));
  }
  __syncthreads();

  // ---- iter 2: d2 = softmax_j(hq*c1); hv = sum_i (d2 + p1)*hm -> sV2 ----
  for (int dslot = 0; dslot < 3; ++dslot) {
    const int d = t + dslot * 256;
    float hqv[8], acc[8];
#pragma unroll
    for (int jj = 0; jj < 8; ++jj) { hqv[jj] = sHq[jj][d]; acc[jj] = 0.f; }
    for (int i = 0; i < I_CAPS; ++i) {
      float av[8], mx = -3.4e38f;
#pragma unroll
      for (int jj = 0; jj < 8; ++jj) { av[jj] = hqv[jj] * sC1[jj * 8 + i]; mx = fmaxf(mx, av[jj]); }
      float se = 0.f;
#pragma unroll
      for (int jj = 0; jj < 8; ++jj) { av[jj] = expf(av[jj] - mx); se += av[jj]; }
      float inv = 1.f / se;
#pragma unroll
      for (int jj = 0; jj < 8; ++jj)
        acc[jj] = fmaf(av[jj] * inv + sP[jj * 8 + i], hm[(jj * 8 + i) * D_DIM + d], acc[jj]);
    }
#pragma unroll
    for (int jj = 0; jj < 8; ++jj) sV2[jj][d] = acc[jj];
  }
  __syncthreads();

  // squash v2, cos2, p2 with y3 = 0.25*hq + 0.25*v1 + 0.5*v2
  {
    float n2 = 0.f;
    for (int d = lane; d < D_DIM; d += 32) { float v = sV2[j][d]; n2 = fmaf(v, v, n2); }
    n2 = wsum32(n2);
    float scale = (n2 / (1.f + n2)) / sqrtf(n2 + EPSF);
    for (int d = lane; d < D_DIM; d += 32) sV2[j][d] *= scale;
  }
  for (int i = 0; i < I_CAPS; ++i) {
    const float* x = hmj + i * D_DIM;
    float dp = 0.f;
    for (int d = lane; d < D_DIM; d += 32) dp = fmaf(x[d], sV2[j][d], dp);
    dp = wsum32(dp);
    if (lane == 0) sC2[j * 8 + i] = dp;
  }
  sum = 0.f; sum2 = 0.f;
  for (int d = lane; d < D_DIM; d += 32) {
    float v = 0.25f * sHq[j][d] + 0.25f * sV1[j][d] + 0.5f * sV2[j][d];
    sum += v; sum2 = fmaf(v, v, sum2);
  }
  sum = wsum32(sum); sum2 = wsum32(sum2);
  meanY = sum * (1.f / 768.f); ssY = sum2 - 768.f * meanY * meanY;
  for (int i = 0; i < I_CAPS; ++i) {
    const float* x = hmj + i * D_DIM;
    float dp = 0.f;
    for (int d = lane; d < D_DIM; d += 32)
      dp = fmaf(x[d], 0.25f * sHq[j][d] + 0.25f * sV1[j][d] + 0.5f * sV2[j][d], dp);
    dp = wsum32(dp);
    float num = dp - 768.f * mstats[j * 8 + i] * meanY;
    float den = sqrtf(mstats[64 + j * 8 + i] * ssY);
    if (lane == 0) sP[j * 8 + i] = tanhf(num / (den + EPSF));
  }
  __syncthreads();

  // ---- final: d3 = softmax_j(hq*c1 + 0.5*(hq+v1)*c2); hv = sum_i (d3+p2)*hm
  for (int dslot = 0; dslot < 3; ++dslot) {
    const int d = t + dslot * 256;
    float hqv[8], ht2[8], acc[8];
#pragma unroll
    for (int jj = 0; jj < 8; ++jj) {
      hqv[jj] = sHq[jj][d];
      ht2[jj] = 0.5f * (hqv[jj] + sV1[jj][d]);
      acc[jj] = 0.f;
    }
    for (int i = 0; i < I_CAPS; ++i) {
      float av[8], mx = -3.4e38f;
#pragma unroll
      for (int jj = 0; jj < 8; ++jj) {
        av[jj] = fmaf(hqv[jj], sC1[jj * 8 + i], ht2[jj] * sC2[jj * 8 + i]);
        mx = fmaxf(mx, av[jj]);
      }
      float se = 0.f;
#pragma unroll
      for (int jj = 0; jj < 8; ++jj) { av[jj] = expf(av[jj] - mx); se += av[jj]; }
      float inv = 1.f / se;
#pragma unroll
      for (int jj = 0; jj < 8; ++jj)
        acc[jj] = fmaf(av[jj] * inv + sP[jj * 8 + i], hm[(jj * 8 + i) * D_DIM + d], acc[jj]);
    }
#pragma unroll
    for (int jj = 0; jj < 8; ++jj) sV2[jj][d] = acc[jj];   // v2 no longer needed
  }
  __syncthreads();

  // squash and write out[n, j*768 + d]
  {
    float n2 = 0.f;
    for (int d = lane; d < D_DIM; d += 32) { float v = sV2[j][d]; n2 = fmaf(v, v, n2); }
    n2 = wsum32(n2);
    float scale = (n2 / (1.f + n2)) / sqrtf(n2 + EPSF);
    float* o = out + (size_t)n * (J_CAPS * D_DIM) + j * D_DIM;
    for (int d = lane; d < D_DIM; d += 32) o[d] = sV2[j][d] * scale;
  }
}

// ---------------------------------------------------------------------------
extern "C" void kernel_launch(void* const* d_in, const int* in_sizes, int n_in,
                              void* d_out, int out_size, void* d_ws, size_t ws_size,
                              hipStream_t stream) {
  const float* m  = (const float*)d_in[0];   // [8,768]
  const float* q  = (const float*)d_in[1];   // [400,768]
  const float* W  = (const float*)d_in[2];   // [1,8,8,768,768] (broadcast over i)
  const float* b  = (const float*)d_in[3];   // [1,8,8,768]
  const float* Wn = (const float*)d_in[4];   // [768,768]
  float* out = (float*)d_out;                // [400, 6144]

  float* ws     = (float*)d_ws;
  float* mproj  = ws;                        // 6144
  float* hm     = ws + 6144;                 // 49152
  float* mstats = ws + 6144 + 49152;         // 128 (mean[64], ss[64])
  float* hq     = ws + 57344;                // 400*8*768 = 2457600

  k_mproj <<<24, 256, 0, stream>>>(m, Wn, mproj);
  k_hatm  <<<192, 256, 0, stream>>>(W, b, mproj, hm);
  k_mstats<<<64, 256, 0, stream>>>(hm, mstats);
  dim3 g(3, 25, 8);
  k_gemm_hq<<<g, 256, 0, stream>>>(q, W, hq);
  k_routing<<<400, 256, 0, stream>>>(hm, mstats, hq, out);
}